// GNN_foraging_complex_71536975282840
// MI455X (gfx1250) — compile-verified
//
#include <hip/hip_runtime.h>

typedef __attribute__((ext_vector_type(2))) float v2f;
typedef __attribute__((ext_vector_type(8))) float v8f;

#define HFEAT 64
#define EPSBN 1e-5f

// ---------------- degree / norm ----------------
__global__ void deg_init(float* deg, int n) {
    int i = blockIdx.x * blockDim.x + threadIdx.x;
    if (i < n) deg[i] = 1.0f;   // self-loop contributes 1
}

__global__ void deg_accum(const long long* __restrict__ dst, float* __restrict__ deg, int E) {
    int e = blockIdx.x * blockDim.x + threadIdx.x;
    if (e < E) atomicAdd(&deg[(int)dst[e]], 1.0f);
}

__global__ void deg_finish(float* deg, int n) {
    int i = blockIdx.x * blockDim.x + threadIdx.x;
    if (i < n) deg[i] = rsqrtf(deg[i]);   // deg >= 1 always
}

// ---------------- fp32 WMMA GEMM: C[M,NC] = A[M,64] @ W[64,NC] (+bias) --------------
// W staged in LDS once per block (<=16KB).  One wave computes a 16xNC strip:
// NC/16 accumulators, so each A operand pair feeds NC/16 WMMAs.
// V_WMMA_F32_16X16X4_F32 layouts (ISA 7.12.2):
//   A 16x4 : lanes 0-15 M=0..15 (K=k0,k0+1), lanes 16-31 M=0..15 (K=k0+2,k0+3)
//   B 4x16 : VGPR0 = K rows {0|2} striped over lanes, VGPR1 = rows {1|3}
//   C 16x16: VGPR v = row v (lanes 0-15) / row v+8 (lanes 16-31), N = lane&15
template<int NC>
__global__ void gemm_wmma_f32(const float* __restrict__ A, const float* __restrict__ W,
                              float* __restrict__ C, const float* __restrict__ bias,
                              int M) {
    constexpr int K = HFEAT;
    __shared__ float sW[K * NC];
    for (int i = threadIdx.x; i < K * NC; i += blockDim.x) sW[i] = W[i];
    __syncthreads();

    int tm   = (blockIdx.x * blockDim.x + threadIdx.x) >> 5;  // one M-tile per wave
    int lane = threadIdx.x & 31;
    if (tm * 16 >= M) return;                // wave-uniform guard: EXEC stays all-1s

    int half = lane >> 4;                    // 0 or 1 (selects K pair)
    int r    = lane & 15;

    v8f acc[NC / 16] = {};
    const float* Arow = A + (size_t)(tm * 16 + r) * K;
#pragma unroll
    for (int k0 = 0; k0 < K; k0 += 4) {
        int ka = k0 + half * 2;
        v2f a;
        a.x = Arow[ka];
        a.y = Arow[ka + 1];
#pragma unroll
        for (int t = 0; t < NC / 16; ++t) {
            v2f b;
            b.x = sW[ka * NC + t * 16 + r];
            b.y = sW[(ka + 1) * NC + t * 16 + r];
            acc[t] = __builtin_amdgcn_wmma_f32_16x16x4_f32(false, a, false, b,
                                                           (short)0, acc[t], false, false);
        }
    }

#pragma unroll
    for (int t = 0; t < NC / 16; ++t) {
        int col = t * 16 + r;
        float bb = bias ? bias[col] : 0.0f;
#pragma unroll
        for (int v = 0; v < 8; ++v) {
            int row = tm * 16 + v + half * 8;
            C[(size_t)row * NC + col] = acc[t][v] + bb;
        }
    }
}

// ---------------- self-loop contribution: Out = H * dinv^2 (per row) ----------------
__global__ void selfloop_init(const float* __restrict__ H, const float* __restrict__ dinv,
                              float* __restrict__ Out, int total) {
    int g = blockIdx.x * blockDim.x + threadIdx.x;
    if (g >= total) return;
    float di = dinv[g >> 6];
    Out[g] = H[g] * di * di;
}

// ---------------- edge scatter: one wave per edge, 2 features per lane ----------------
__global__ void edge_scatter(const float* __restrict__ H, const float* __restrict__ dinv,
                             const long long* __restrict__ src, const long long* __restrict__ dst,
                             float* __restrict__ Out, int E) {
    int wid  = (blockIdx.x * blockDim.x + threadIdx.x) >> 5;
    int lane = threadIdx.x & 31;
    if (wid >= E) return;
    int s = (int)src[wid];
    int d = (int)dst[wid];
    float norm = dinv[s] * dinv[d];            // wave-uniform -> scalarized
    const float2* hs = (const float2*)(H + (size_t)s * HFEAT);
    float2 v = hs[lane];
    float* o = Out + (size_t)d * HFEAT + lane * 2;
    atomicAdd(o,     v.x * norm);
    atomicAdd(o + 1, v.y * norm);
}

// ---------------- zero the BN stats accumulators ----------------
__global__ void zero128(float* p) {
    if (threadIdx.x < 128) p[threadIdx.x] = 0.0f;
}

// ---------------- fused bias + ReLU + BN statistics (sum, sumsq per column) ----------
__global__ void bias_relu_stats(float* __restrict__ X, const float* __restrict__ bias,
                                float* __restrict__ sum, float* __restrict__ sumsq, int total) {
    __shared__ float sS[HFEAT], sQ[HFEAT];
    int tid = threadIdx.x;
    if (tid < HFEAT) { sS[tid] = 0.0f; sQ[tid] = 0.0f; }
    __syncthreads();

    int g = blockIdx.x * blockDim.x + tid;
    int stride = gridDim.x * blockDim.x;       // multiple of 64 -> col fixed per thread
    int col = g & (HFEAT - 1);
    float b = bias[col];
    float ls = 0.0f, lq = 0.0f;
    for (int i = g; i < total; i += stride) {
        float v = X[i] + b;
        v = v > 0.0f ? v : 0.0f;
        X[i] = v;
        ls += v;
        lq += v * v;
    }
    atomicAdd(&sS[col], ls);                   // LDS ds_add_f32
    atomicAdd(&sQ[col], lq);
    __syncthreads();
    if (tid < HFEAT) {
        atomicAdd(&sum[tid],   sS[tid]);
        atomicAdd(&sumsq[tid], sQ[tid]);
    }
}

// ---------------- BN apply ----------------
__global__ void bn_apply(float* __restrict__ X, const float* __restrict__ sum,
                         const float* __restrict__ sumsq, const float* __restrict__ gamma,
                         const float* __restrict__ beta, int total, float invN) {
    int g = blockIdx.x * blockDim.x + threadIdx.x;
    int stride = gridDim.x * blockDim.x;
    int col = g & (HFEAT - 1);
    float mu  = sum[col] * invN;
    float var = sumsq[col] * invN - mu * mu;
    float sc  = rsqrtf(var + EPSBN) * gamma[col];
    float bt  = beta[col];
    for (int i = g; i < total; i += stride) {
        X[i] = (X[i] - mu) * sc + bt;
    }
}

extern "C" void kernel_launch(void* const* d_in, const int* in_sizes, int n_in,
                              void* d_out, int out_size, void* d_ws, size_t ws_size,
                              hipStream_t stream) {
    const float*     x   = (const float*)d_in[0];
    const long long* ei  = (const long long*)d_in[1];
    const float*     W1  = (const float*)d_in[2];
    const float*     b1  = (const float*)d_in[3];
    const float*     g1  = (const float*)d_in[4];
    const float*     be1 = (const float*)d_in[5];
    const float*     W2  = (const float*)d_in[6];
    const float*     b2  = (const float*)d_in[7];
    const float*     g2  = (const float*)d_in[8];
    const float*     be2 = (const float*)d_in[9];
    const float*     Wfc = (const float*)d_in[10];
    const float*     bfc = (const float*)d_in[11];

    const int n = in_sizes[0] / HFEAT;         // 100000
    const int E = in_sizes[1] / 2;             // 1600000
    const int total = n * HFEAT;

    // workspace layout (256-element aligned segments)
    float* dinv  = (float*)d_ws;
    float* bufA  = dinv + ((n + 255) / 256) * 256;   // N x 64
    float* bufB  = bufA + (size_t)n * HFEAT;         // N x 64
    float* stats = bufB + (size_t)n * HFEAT;         // sum[64], sumsq[64]

    const long long* src = ei;
    const long long* dst = ei + E;

    const int TB = 256;
    dim3 blk(TB);

    // --- degree / symmetric norm (shared by both layers) ---
    deg_init  <<<dim3((n + TB - 1) / TB), blk, 0, stream>>>(dinv, n);
    deg_accum <<<dim3((E + TB - 1) / TB), blk, 0, stream>>>(dst, dinv, E);
    deg_finish<<<dim3((n + TB - 1) / TB), blk, 0, stream>>>(dinv, n);

    const int mtiles = (n + 15) / 16;                 // one M-tile per wave
    const unsigned gemm_grid = (unsigned)((mtiles + (TB / 32) - 1) / (TB / 32));
    dim3 gridE((E + (TB / 32) - 1) / (TB / 32));
    dim3 gridT((total + TB - 1) / TB);
    dim3 gridS(512);

    // --- layer 1 ---
    gemm_wmma_f32<HFEAT><<<dim3(gemm_grid), blk, 0, stream>>>(x, W1, bufA, nullptr, n);
    selfloop_init<<<gridT, blk, 0, stream>>>(bufA, dinv, bufB, total);
    edge_scatter <<<gridE, blk, 0, stream>>>(bufA, dinv, src, dst, bufB, E);
    zero128      <<<dim3(1), dim3(128), 0, stream>>>(stats);
    bias_relu_stats<<<gridS, blk, 0, stream>>>(bufB, b1, stats, stats + HFEAT, total);
    bn_apply     <<<gridS, blk, 0, stream>>>(bufB, stats, stats + HFEAT, g1, be1, total, 1.0f / n);

    // --- layer 2 ---
    gemm_wmma_f32<HFEAT><<<dim3(gemm_grid), blk, 0, stream>>>(bufB, W2, bufA, nullptr, n);
    selfloop_init<<<gridT, blk, 0, stream>>>(bufA, dinv, bufB, total);
    edge_scatter <<<gridE, blk, 0, stream>>>(bufA, dinv, src, dst, bufB, E);
    zero128      <<<dim3(1), dim3(128), 0, stream>>>(stats);
    bias_relu_stats<<<gridS, blk, 0, stream>>>(bufB, b2, stats, stats + HFEAT, total);
    bn_apply     <<<gridS, blk, 0, stream>>>(bufB, stats, stats + HFEAT, g2, be2, total, 1.0f / n);

    // --- final FC: d_out = bufB @ Wfc + bfc  (N x 32) ---
    gemm_wmma_f32<32><<<dim3(gemm_grid), blk, 0, stream>>>(bufB, Wfc, (float*)d_out, bfc, n);
}